// LocalSpearmanCorrLoss_13245679141000
// MI455X (gfx1250) — compile-verified
//
#include <hip/hip_runtime.h>
#include <stdint.h>

#ifndef __has_builtin
#define __has_builtin(x) 0
#endif

#define ROW_N   1024
#define THREADS 256

typedef __attribute__((ext_vector_type(2))) float v2f;
typedef __attribute__((ext_vector_type(8))) float v8f;
typedef __attribute__((__vector_size__(4 * sizeof(int)))) int v4i;

#if __has_builtin(__builtin_amdgcn_global_load_async_to_lds_b128) && \
    __has_builtin(__builtin_amdgcn_s_wait_asynccnt)
#define USE_ASYNC_LDS 1
// global (AS1) and LDS (AS3) qualified vector pointers for the async builtin
typedef __attribute__((address_space(1))) v4i gas_v4i;
typedef __attribute__((address_space(3))) v4i las_v4i;
#endif

// Monotonic key: bigger float -> smaller u32 key (ascending u64 sort == stable descending argsort)
__device__ __forceinline__ uint32_t key_desc_from_f32(float f) {
  uint32_t u = __float_as_uint(f);
  uint32_t m = u ^ ((u >> 31) ? 0xFFFFFFFFu : 0x80000000u);  // order-preserving map
  return ~m;                                                  // invert for descending
}
__device__ __forceinline__ float f32_from_key_desc(uint32_t k) {
  uint32_t m = ~k;
  uint32_t u = (m & 0x80000000u) ? (m ^ 0x80000000u) : ~m;
  return __uint_as_float(u);
}

struct Smem {
  unsigned long long keyP[ROW_N];   // (desc key << 32) | index
  unsigned long long keyT[ROW_N];
  float sValP[ROW_N];               // staging, then sorted values
  float sValT[ROW_N];
  float stValP[ROW_N]; float stSumP[ROW_N]; float stCntP[ROW_N];  // PAVA stacks
  float stValT[ROW_N]; float stSumT[ROW_N]; float stCntT[ROW_N];
  float rp[ROW_N]; float rt[ROW_N]; // soft ranks in original order
  float red[5 * 32];                // cross-lane reduction scratch
};

__global__ __launch_bounds__(THREADS)
void spearman_rows(const float* __restrict__ pred, const float* __restrict__ target,
                   float* __restrict__ row_loss, int n_rows)
{
  __shared__ Smem sm;
  const int row = blockIdx.x;
  if (row >= n_rows) return;                     // uniform per block
  const int tid = threadIdx.x;
  const float* prow = pred   + (size_t)row * ROW_N;
  const float* trow = target + (size_t)row * ROW_N;

  __builtin_prefetch(trow + tid * 4, 0, 1);      // gfx1250 global_prefetch

  // ---- Stage rows into LDS and pack sort keys ----
#if defined(USE_ASYNC_LDS)
  {
    __builtin_amdgcn_global_load_async_to_lds_b128(
        (gas_v4i*)(prow + tid * 4), (las_v4i*)&sm.sValP[tid * 4], 0, 0);
    __builtin_amdgcn_global_load_async_to_lds_b128(
        (gas_v4i*)(trow + tid * 4), (las_v4i*)&sm.sValT[tid * 4], 0, 0);
    __builtin_amdgcn_s_wait_asynccnt(0);
  }
  __syncthreads();
  for (int i = tid; i < ROW_N; i += THREADS) {
    sm.keyP[i] = (((unsigned long long)key_desc_from_f32(sm.sValP[i])) << 32) | (unsigned)i;
    sm.keyT[i] = (((unsigned long long)key_desc_from_f32(sm.sValT[i])) << 32) | (unsigned)i;
  }
#else
  for (int i = tid; i < ROW_N; i += THREADS) {
    sm.keyP[i] = (((unsigned long long)key_desc_from_f32(prow[i])) << 32) | (unsigned)i;
    sm.keyT[i] = (((unsigned long long)key_desc_from_f32(trow[i])) << 32) | (unsigned)i;
  }
#endif
  __syncthreads();

  // ---- Bitonic sort (ascending u64 == stable descending-by-value) on both arrays ----
  for (int k = 2; k <= ROW_N; k <<= 1) {
    for (int j = k >> 1; j > 0; j >>= 1) {
#pragma unroll
      for (int base = 0; base < ROW_N; base += THREADS) {
        int i   = base + tid;
        int ixj = i ^ j;
        if (ixj > i) {   // each pair owned by exactly one thread per stage
          bool up = ((i & k) == 0);
          unsigned long long a = sm.keyP[i], b = sm.keyP[ixj];
          if ((a > b) == up) { sm.keyP[i] = b; sm.keyP[ixj] = a; }
          unsigned long long c = sm.keyT[i], d = sm.keyT[ixj];
          if ((c > d) == up) { sm.keyT[i] = d; sm.keyT[ixj] = c; }
        }
      }
      __syncthreads();
    }
  }

  // ---- Unpack sorted values ----
  for (int i = tid; i < ROW_N; i += THREADS) {
    sm.sValP[i] = f32_from_key_desc((uint32_t)(sm.keyP[i] >> 32));
    sm.sValT[i] = f32_from_key_desc((uint32_t)(sm.keyT[i] >> 32));
  }
  __syncthreads();

  // ---- PAVA (decreasing isotonic) + expansion + scatter; pred on wave0 lane0,
  //      target on wave1 lane0 so the two serial passes overlap on different SIMDs ----
  if (tid == 0 || tid == 32) {
    const bool isP = (tid == 0);
    const float* sVal = isP ? sm.sValP : sm.sValT;
    const unsigned long long* key = isP ? sm.keyP : sm.keyT;
    float* stV  = isP ? sm.stValP : sm.stValT;
    float* stS  = isP ? sm.stSumP : sm.stSumT;
    float* stC  = isP ? sm.stCntP : sm.stCntT;
    float* rOut = isP ? sm.rp     : sm.rt;

    int sp = 0;
    for (int i = 0; i < ROW_N; ++i) {
      float v = sVal[i] - (float)(ROW_N - i);    // y_i = s_i - w_i, w = n..1
      float s = v, c = 1.0f;
      while (sp > 0 && stV[sp - 1] <= v) {       // merge while prev block <= current
        s += stS[sp - 1];
        c += stC[sp - 1];
        v = s / c;
        --sp;
      }
      stV[sp] = v; stS[sp] = s; stC[sp] = c; ++sp;
    }
    // expand blocks: ranks_sorted = s - sol; scatter through inverse permutation
    int pos = 0;
    for (int b = 0; b < sp; ++b) {
      float v = stV[b];
      int   c = (int)stC[b];
      for (int e = 0; e < c; ++e, ++pos) {
        int orig = (int)(unsigned)(key[pos] & 0xFFFFFFFFull);
        rOut[orig] = sVal[pos] - v;
      }
    }
  }
  __syncthreads();

  // ---- Row statistics via full-fp32 WMMA (V_WMMA_F32_16X16X4_F32).
  //      A rows / B cols hold the same 16 blocks of 4 elements, so the diagonal of
  //      D = A*B + C accumulates exact per-block dot products; B=ones gives sums. ----
#if __has_builtin(__builtin_amdgcn_wmma_f32_16x16x4_f32)
  if (tid < 32) {
    const int lane = tid;
    // A 16x4 f32 layout: lanes 0-15 -> (M=lane, K=0/1), lanes 16-31 -> (M=lane-16, K=2/3).
    // B 4x16 mirrors it, so element index = blk*4 + k with blk = lane&15 for both roles.
    const int idx0 = ((lane & 15) << 2) + ((lane >> 4) << 1);
    v8f cP = {}, cT = {}, cPT = {}, cPP = {}, cTT = {};
    v2f ones; ones.x = 1.0f; ones.y = 1.0f;
#pragma unroll
    for (int t = 0; t < 16; ++t) {
      const int base = t * 64 + idx0;
      v2f p; p.x = sm.rp[base]; p.y = sm.rp[base + 1];
      v2f q; q.x = sm.rt[base]; q.y = sm.rt[base + 1];
      cP  = __builtin_amdgcn_wmma_f32_16x16x4_f32(false, p, false, ones, (short)0, cP,  false, false);
      cT  = __builtin_amdgcn_wmma_f32_16x16x4_f32(false, q, false, ones, (short)0, cT,  false, false);
      cPT = __builtin_amdgcn_wmma_f32_16x16x4_f32(false, p, false, q,    (short)0, cPT, false, false);
      cPP = __builtin_amdgcn_wmma_f32_16x16x4_f32(false, p, false, p,    (short)0, cPP, false, false);
      cTT = __builtin_amdgcn_wmma_f32_16x16x4_f32(false, q, false, q,    (short)0, cTT, false, false);
    }
    // Diagonal lives at: lanes 0-7 vgpr r=lane (M=N=r), lanes 24-31 vgpr r=lane-24 (M=N=8+r).
    const int  r  = lane & 7;
    const float m = (lane < 8 || lane >= 24) ? 1.0f : 0.0f;
    sm.red[0 * 32 + lane] = cP[r]  * m;
    sm.red[1 * 32 + lane] = cT[r]  * m;
    sm.red[2 * 32 + lane] = cPT[r] * m;
    sm.red[3 * 32 + lane] = cPP[r] * m;
    sm.red[4 * 32 + lane] = cTT[r] * m;
  }
#else
  if (tid < 32) {
    float sP = 0.f, sT = 0.f, sPT = 0.f, sPP = 0.f, sTT = 0.f;
    for (int i = tid; i < ROW_N; i += 32) {
      float p = sm.rp[i], q = sm.rt[i];
      sP += p; sT += q; sPT += p * q; sPP += p * p; sTT += q * q;
    }
    sm.red[0 * 32 + tid] = sP;
    sm.red[1 * 32 + tid] = sT;
    sm.red[2 * 32 + tid] = sPT;
    sm.red[3 * 32 + tid] = sPP;
    sm.red[4 * 32 + tid] = sTT;
  }
#endif
  __syncthreads();

  if (tid == 0) {
    float S[5];
#pragma unroll
    for (int s = 0; s < 5; ++s) {
      float acc = 0.f;
      for (int l = 0; l < 32; ++l) acc += sm.red[s * 32 + l];
      S[s] = acc;
    }
    const float n   = (float)ROW_N;
    const float dot = S[2] - S[0] * S[1] / n;            // centered dot
    const float pp  = S[3] - S[0] * S[0] / n;            // centered ||p||^2
    const float tt  = S[4] - S[1] * S[1] / n;            // centered ||t||^2
    const float den = fmaxf(sqrtf(fmaxf(pp, 0.f)) * sqrtf(fmaxf(tt, 0.f)), 1e-8f);
    row_loss[row] = 1.0f - dot / den;
  }
}

// Deterministic fixed-order mean over n_rows row losses.
__global__ __launch_bounds__(256)
void mean_rows(const float* __restrict__ row_loss, float* __restrict__ out, int n_rows)
{
  __shared__ float buf[256];
  const int tid = threadIdx.x;
  float acc = 0.f;
  for (int i = tid; i < n_rows; i += 256) acc += row_loss[i];
  buf[tid] = acc;
  __syncthreads();
  for (int s = 128; s > 0; s >>= 1) {
    if (tid < s) buf[tid] += buf[tid + s];
    __syncthreads();
  }
  if (tid == 0) out[0] = buf[0] / (float)n_rows;
}

extern "C" void kernel_launch(void* const* d_in, const int* in_sizes, int n_in,
                              void* d_out, int out_size, void* d_ws, size_t ws_size,
                              hipStream_t stream) {
  const float* pred   = (const float*)d_in[0];
  const float* target = (const float*)d_in[1];
  float* out      = (float*)d_out;
  float* row_loss = (float*)d_ws;              // n_rows floats of scratch
  const int n_rows = in_sizes[0] / ROW_N;      // 8192

  spearman_rows<<<n_rows, THREADS, 0, stream>>>(pred, target, row_loss, n_rows);
  mean_rows<<<1, 256, 0, stream>>>(row_loss, out, n_rows);
}